// GRU_83021717831946
// MI455X (gfx1250) — compile-verified
//
#include <hip/hip_runtime.h>
#include <math.h>

typedef __attribute__((ext_vector_type(2))) float v2f;
typedef __attribute__((ext_vector_type(8))) float v8f;

#define T_STEPS 1024
#define BATCH   64
#define HID     512
#define G3      1536      // 3*H
#define NWG     32        // one workgroup per 16-wide column tile of H

__device__ __forceinline__ float sigmoidf_(float x) {
    return 1.0f / (1.0f + __expf(-x));
}

// ---------------------------------------------------------------------------
// Phase 1: gx = A[T*B, 512] @ W[1536, 512]^T + bias   ->  gx[T*B, 1536]
// One wave = one 16x16 M-tile x four 16-wide N-tiles (A-fragment reuse).
// ---------------------------------------------------------------------------
__global__ __launch_bounds__(256) void gx_gemm(const float* __restrict__ A,
                                               const float* __restrict__ W,
                                               const float* __restrict__ bias,
                                               float* __restrict__ gx) {
    const int lane = threadIdx.x & 31;
    const int wave = threadIdx.x >> 5;
    const int job  = blockIdx.x * 8 + wave;   // 98304 wave-jobs total
    const int tileM = job / 24;               // 4096 M-tiles (T*B/16)
    const int nb    = job % 24;               // 24 groups of 4 N-tiles
    const int m0 = tileM * 16;
    const int n0 = nb * 64;

    const int ml   = lane & 15;               // A: row m / B: col n
    const int koff = (lane >> 4) * 2;         // K half select

    v8f acc0 = {}, acc1 = {}, acc2 = {}, acc3 = {};
    const float* arow = A + (size_t)(m0 + ml) * 512 + koff;
    const float* w0 = W + (size_t)(n0 +  0 + ml) * 512 + koff;
    const float* w1 = W + (size_t)(n0 + 16 + ml) * 512 + koff;
    const float* w2 = W + (size_t)(n0 + 32 + ml) * 512 + koff;
    const float* w3 = W + (size_t)(n0 + 48 + ml) * 512 + koff;

#pragma unroll 8
    for (int k = 0; k < 512; k += 4) {
        v2f a  = *(const v2f*)(arow + k);
        v2f b0 = *(const v2f*)(w0 + k);
        v2f b1 = *(const v2f*)(w1 + k);
        v2f b2 = *(const v2f*)(w2 + k);
        v2f b3 = *(const v2f*)(w3 + k);
        acc0 = __builtin_amdgcn_wmma_f32_16x16x4_f32(false, a, false, b0, (short)0, acc0, false, false);
        acc1 = __builtin_amdgcn_wmma_f32_16x16x4_f32(false, a, false, b1, (short)0, acc1, false, false);
        acc2 = __builtin_amdgcn_wmma_f32_16x16x4_f32(false, a, false, b2, (short)0, acc2, false, false);
        acc3 = __builtin_amdgcn_wmma_f32_16x16x4_f32(false, a, false, b3, (short)0, acc3, false, false);
    }

    const int rbase = m0 + (lane >> 4) * 8;   // C/D layout: M = v + 8*(lane>=16)
    const int cl = lane & 15;
#pragma unroll
    for (int v = 0; v < 8; ++v) {
        float* o = gx + (size_t)(rbase + v) * G3 + n0 + cl;
        o[0]  = acc0[v] + bias[n0 +  0 + cl];
        o[16] = acc1[v] + bias[n0 + 16 + cl];
        o[32] = acc2[v] + bias[n0 + 32 + cl];
        o[48] = acc3[v] + bias[n0 + 48 + cl];
    }
}

// ---------------------------------------------------------------------------
// init: h ping buffer <- hx[layer], zero the step counter
// ---------------------------------------------------------------------------
__global__ void init_h(const float* __restrict__ hx, float* __restrict__ h0,
                       unsigned int* __restrict__ counter) {
    int i = blockIdx.x * blockDim.x + threadIdx.x;
    if (i < BATCH * HID) h0[i] = hx[i];
    if (i == 0) *counter = 0u;
}

// ---------------------------------------------------------------------------
// Phase 2: persistent recurrent scan. Grid = 32 WGs x 128 threads (4 waves).
// WG b owns h columns [16b, 16b+16); wave w owns batch rows [16w, 16w+16).
// w_hh slice (3 gates x 16 rows x 512) lives in LDS, rows padded to 514
// floats for conflict-free fragment reads. Hidden state double-buffered in
// global memory; one device-scope barrier per timestep.
// ---------------------------------------------------------------------------
__global__ __launch_bounds__(128) void gru_scan(const float* __restrict__ gx,   // [T,B,1536] incl b_ih
                                                const float* __restrict__ whh,  // [1536,512]
                                                const float* __restrict__ bhh,  // [1536]
                                                float* __restrict__ hping,      // [B*H]
                                                float* __restrict__ hpong,      // [B*H]
                                                float* __restrict__ out,        // [T,B,H]
                                                float* __restrict__ hfin,       // [B*H]
                                                unsigned int* __restrict__ counter) {
    extern __shared__ float lds[];            // 3 * 16 * 514 floats
    const int n0   = blockIdx.x * 16;
    const int lane = threadIdx.x & 31;
    const int wave = threadIdx.x >> 5;
    const int m0   = wave * 16;

    // Preload weight slice: lds[g*16*514 + r*514 + k] = whh[g*512 + n0 + r][k]
    for (int idx = threadIdx.x; idx < 3 * 16 * 256; idx += 128) {
        int g  = idx / (16 * 256);
        int r  = (idx / 256) & 15;
        int k2 = (idx & 255) * 2;
        v2f w = *(const v2f*)(whh + (size_t)(g * 512 + n0 + r) * 512 + k2);
        *(v2f*)(lds + g * (16 * 514) + r * 514 + k2) = w;
    }
    __syncthreads();

    const int ml   = lane & 15;
    const int koff = (lane >> 4) * 2;
    const float* wg0 = lds + 0 * (16 * 514) + ml * 514 + koff;
    const float* wg1 = lds + 1 * (16 * 514) + ml * 514 + koff;
    const float* wg2 = lds + 2 * (16 * 514) + ml * 514 + koff;

    const int rbase = m0 + (lane >> 4) * 8;   // C rows this lane holds
    const int col   = n0 + ml;                // C column this lane holds
    const float bh_r = bhh[0 * 512 + col];
    const float bh_z = bhh[1 * 512 + col];
    const float bh_n = bhh[2 * 512 + col];

    for (int t = 0; t < T_STEPS; ++t) {
        const float* hin  = (t & 1) ? hpong : hping;
        float*       hout = (t & 1) ? hping : hpong;

        v8f ar = {}, az = {}, an = {};
        const float* hrow = hin + (size_t)(m0 + ml) * 512 + koff;
#pragma unroll 4
        for (int k = 0; k < 512; k += 4) {
            v2f a  = *(const v2f*)(hrow + k);
            v2f br = *(const v2f*)(wg0 + k);
            v2f bz = *(const v2f*)(wg1 + k);
            v2f bn = *(const v2f*)(wg2 + k);
            ar = __builtin_amdgcn_wmma_f32_16x16x4_f32(false, a, false, br, (short)0, ar, false, false);
            az = __builtin_amdgcn_wmma_f32_16x16x4_f32(false, a, false, bz, (short)0, az, false, false);
            an = __builtin_amdgcn_wmma_f32_16x16x4_f32(false, a, false, bn, (short)0, an, false, false);
        }

        const float* gxt = gx + (size_t)t * BATCH * G3;
#pragma unroll
        for (int v = 0; v < 8; ++v) {
            int row = rbase + v;
            const float* g = gxt + (size_t)row * G3;
            float i_r = g[0 * 512 + col];
            float i_z = g[1 * 512 + col];
            float i_n = g[2 * 512 + col];
            float r  = sigmoidf_(i_r + ar[v] + bh_r);
            float z  = sigmoidf_(i_z + az[v] + bh_z);
            float nn = tanhf(i_n + r * (an[v] + bh_n));
            float hprev = hin[(size_t)row * 512 + col];
            float hy = nn + z * (hprev - nn);
            hout[(size_t)row * 512 + col] = hy;
            out[(size_t)t * BATCH * HID + (size_t)row * HID + col] = hy;
            if (t == T_STEPS - 1) hfin[(size_t)row * HID + col] = hy;
        }

        // ---- device-scope split barrier (one per step) ----
        __builtin_amdgcn_fence(__ATOMIC_RELEASE, "agent");
        __syncthreads();
        if (threadIdx.x == 0) {
            __hip_atomic_fetch_add(counter, 1u, __ATOMIC_RELEASE, __HIP_MEMORY_SCOPE_AGENT);
            unsigned int target = (unsigned int)(NWG * (t + 1));
            while (__hip_atomic_load(counter, __ATOMIC_ACQUIRE, __HIP_MEMORY_SCOPE_AGENT) < target) {
                __builtin_amdgcn_s_sleep(8);
            }
        }
        __syncthreads();
        __builtin_amdgcn_fence(__ATOMIC_ACQUIRE, "agent");
    }
}

// ---------------------------------------------------------------------------
extern "C" void kernel_launch(void* const* d_in, const int* in_sizes, int n_in,
                              void* d_out, int out_size, void* d_ws, size_t ws_size,
                              hipStream_t stream) {
    const float* x    = (const float*)d_in[0];
    const float* hx   = (const float*)d_in[1];
    const float* wih0 = (const float*)d_in[2];
    const float* whh0 = (const float*)d_in[3];
    const float* bih0 = (const float*)d_in[4];
    const float* bhh0 = (const float*)d_in[5];
    const float* wih1 = (const float*)d_in[6];
    const float* whh1 = (const float*)d_in[7];
    const float* bih1 = (const float*)d_in[8];
    const float* bhh1 = (const float*)d_in[9];

    float* out1 = (float*)d_out;                                   // [T,B,H]
    float* hst  = (float*)d_out + (size_t)T_STEPS * BATCH * HID;   // [2,B,H]

    float* ws   = (float*)d_ws;
    float* gx   = ws;                                              // T*B*1536
    float* out0 = gx + (size_t)T_STEPS * BATCH * G3;               // T*B*512
    float* hA   = out0 + (size_t)T_STEPS * BATCH * HID;            // B*H
    float* hB   = hA + (size_t)BATCH * HID;                        // B*H
    unsigned int* counter = (unsigned int*)(hB + (size_t)BATCH * HID);

    const dim3 ggrd(12288), gblk(256);       // 4096 M-tiles * 24 N-jobs / 8 waves
    const size_t ldsBytes = 3 * 16 * 514 * sizeof(float);

    // ---- layer 0 ----
    init_h<<<128, 256, 0, stream>>>(hx, hA, counter);
    gx_gemm<<<ggrd, gblk, 0, stream>>>(x, wih0, bih0, gx);
    gru_scan<<<NWG, 128, ldsBytes, stream>>>(gx, whh0, bhh0, hA, hB, out0, hst, counter);

    // ---- layer 1 ----
    init_h<<<128, 256, 0, stream>>>(hx + (size_t)BATCH * HID, hA, counter);
    gx_gemm<<<ggrd, gblk, 0, stream>>>(out0, wih1, bih1, gx);
    gru_scan<<<NWG, 128, ldsBytes, stream>>>(gx, whh1, bhh1, hA, hB, out1,
                                             hst + (size_t)BATCH * HID, counter);
}